// TSPAttentionCriterion_56951266345756
// MI455X (gfx1250) — compile-verified
//
#include <hip/hip_runtime.h>
#include <math.h>

#define S_DIM 64
#define B_DIM 128
#define E_DIM 2000
#define N_DIM 100
#define BLOCK 256
#define NTILE ((E_DIM + BLOCK - 1) / BLOCK) /* 8 */

__global__ __launch_bounds__(BLOCK)
void tsp_bce_kernel(const float* __restrict__ output,
                    const int*   __restrict__ targets,
                    const int*   __restrict__ edges,
                    const int*   __restrict__ edge_counts,
                    float*       __restrict__ losses)
{
    // Successor table for this (s,b) tour: succ[tour[k]] = tour[(k+1)%N]
    __shared__ int succ[N_DIM + 28];                 // pad to avoid bank aliasing
    // Double-buffered async staging: per-slot 8B edge pair + 4B logit
    __shared__ __align__(16) int   ebuf[2][BLOCK][2];
    __shared__ __align__(16) float xbuf[2][BLOCK];
    __shared__ float partial[BLOCK / 32];

    const int tid = threadIdx.x;
    const int blk = blockIdx.x;                      // 0 .. S*B-1

    // ---- build successor table (threads 0..99) ----
    const int* tgt = targets + (size_t)blk * N_DIM;
    if (tid < N_DIM) {
        int a  = tgt[tid];
        int nb = tgt[(tid + 1 == N_DIM) ? 0 : tid + 1];
        succ[a] = nb;
    }

    const int cnt = edge_counts[blk];
    const unsigned long long e_base =
        (unsigned long long)(const void*)edges  + (unsigned long long)blk * ((unsigned long long)E_DIM * 8ull);
    const unsigned long long x_base =
        (unsigned long long)(const void*)output + (unsigned long long)blk * ((unsigned long long)E_DIM * 4ull);

    // ---- prologue: async-prefetch tile 0 into buffer 0 (ASYNCcnt path) ----
    {
        unsigned le = (unsigned)(size_t)&ebuf[0][tid][0];   // low 32 bits of a generic
        unsigned lx = (unsigned)(size_t)&xbuf[0][tid];      // shared ptr == LDS offset
        unsigned ve = (unsigned)(tid * 8);
        unsigned vx = (unsigned)(tid * 4);
        asm volatile("global_load_async_to_lds_b64 %0, %1, %2"
                     :: "v"(le), "v"(ve), "s"(e_base) : "memory");
        asm volatile("global_load_async_to_lds_b32 %0, %1, %2"
                     :: "v"(lx), "v"(vx), "s"(x_base) : "memory");
    }

    __syncthreads();    // succ table visible to whole block

    float sum = 0.0f;
    for (int k = 0; k < NTILE; ++k) {
        const int buf = k & 1;
        if (k + 1 < NTILE) {
            // Guard LDS WAR: our previous reads of slot buf^1 must retire
            // before the async engine can overwrite it.
            if (k > 0) asm volatile("s_wait_dscnt 0" ::: "memory");
            int idx1 = (k + 1) * BLOCK + tid;
            int g    = (idx1 < E_DIM) ? idx1 : (E_DIM - 1);   // clamp tail, masked later
            unsigned le = (unsigned)(size_t)&ebuf[buf ^ 1][tid][0];
            unsigned lx = (unsigned)(size_t)&xbuf[buf ^ 1][tid];
            unsigned ve = (unsigned)(g * 8);
            unsigned vx = (unsigned)(g * 4);
            asm volatile("global_load_async_to_lds_b64 %0, %1, %2"
                         :: "v"(le), "v"(ve), "s"(e_base) : "memory");
            asm volatile("global_load_async_to_lds_b32 %0, %1, %2"
                         :: "v"(lx), "v"(vx), "s"(x_base) : "memory");
            // 2 newer ops in flight (tile k+1); wait until tile k's 2 are done.
            asm volatile("s_wait_asynccnt 2" ::: "memory");
        } else {
            asm volatile("s_wait_asynccnt 0" ::: "memory");
        }

        const int   idx = k * BLOCK + tid;
        const int   u   = ebuf[buf][tid][0];
        const int   v   = ebuf[buf][tid][1];
        const float x   = xbuf[buf][tid];

        const bool lab   = (succ[u] == v) || (succ[v] == u);
        const bool valid = (idx < cnt) && (idx < E_DIM);
        const float z = lab ? 1.0f : 0.0f;
        const float w = lab ? 1.0f : 0.25f;
        // stable BCE-with-logits: max(x,0) - x*z + log1p(exp(-|x|))
        const float bce = fmaxf(x, 0.0f) - x * z + log1pf(expf(-fabsf(x)));
        if (valid) sum += bce * w;
    }

    // ---- wave32 reduction, then cross-wave combine ----
    for (int off = 16; off > 0; off >>= 1)
        sum += __shfl_down(sum, off, 32);
    if ((tid & 31) == 0) partial[tid >> 5] = sum;
    __syncthreads();
    if (tid == 0) {
        float t = 0.0f;
#pragma unroll
        for (int i = 0; i < BLOCK / 32; ++i) t += partial[i];
        losses[blk] = t * (1.0f / (float)N_DIM);
    }
}

extern "C" void kernel_launch(void* const* d_in, const int* in_sizes, int n_in,
                              void* d_out, int out_size, void* d_ws, size_t ws_size,
                              hipStream_t stream) {
    const float* output      = (const float*)d_in[0];
    const int*   targets     = (const int*)d_in[1];
    const int*   edges       = (const int*)d_in[2];
    const int*   edge_counts = (const int*)d_in[3];
    // d_in[4] = single_eval_pos (unused by the reference math)
    float* losses = (float*)d_out;

    dim3 grid(S_DIM * B_DIM);
    dim3 block(BLOCK);
    hipLaunchKernelGGL(tsp_bce_kernel, grid, block, 0, stream,
                       output, targets, edges, edge_counts, losses);
}